// DAA_66812511256800
// MI455X (gfx1250) — compile-verified
//
#include <hip/hip_runtime.h>
#include <hip/hip_bf16.h>

typedef __attribute__((ext_vector_type(2))) float v2f;
typedef __attribute__((ext_vector_type(8))) float v8f;

#define M_TOT 320          // N*K = 64*5
#define D_DIM 512
#define N_IMG 64
#define K_CAP 5
#define INV_T 100.0f       // 1/0.01

// ---------------------------------------------------------------------------
// Kernel 1: score = A * A^T  (320x320, f32) via V_WMMA_F32_16X16X4_F32.
// One wave32 per 16x16 output tile; K-loop over 512 in steps of 4.
// A-frag (16x4 MxK): lane L -> row = L%16, k-pair base = 2*(L/16)  (float2 load)
// B-frag (4x16 KxN): B[k,n] = A[colBase+n, k] -> identical per-lane addressing
// with the column tile base (score is A*A^T).
// ---------------------------------------------------------------------------
__global__ __launch_bounds__(32) void daa_score_gemm(const float* __restrict__ A,
                                                     float* __restrict__ S) {
  const int bm   = blockIdx.x;
  const int bn   = blockIdx.y;
  const int lane = threadIdx.x;
  const int half = lane >> 4;   // 0 or 1
  const int l    = lane & 15;

  const float* __restrict__ arow = A + (size_t)(bm * 16 + l) * D_DIM + 2 * half;
  const float* __restrict__ brow = A + (size_t)(bn * 16 + l) * D_DIM + 2 * half;

  v8f c = {0.f, 0.f, 0.f, 0.f, 0.f, 0.f, 0.f, 0.f};

#pragma unroll 4
  for (int k = 0; k < D_DIM; k += 4) {
    v2f a = *(const v2f*)(arow + k);
    v2f b = *(const v2f*)(brow + k);
    // 8 args: (neg_a, A, neg_b, B, c_mod, C, reuse_a, reuse_b)
    c = __builtin_amdgcn_wmma_f32_16x16x4_f32(false, a, false, b,
                                              (short)0, c, false, false);
  }

  // D layout: VGPR v -> M = v (lanes 0-15) / v+8 (lanes 16-31), N = lane%16
  float* __restrict__ out = S + (size_t)(bm * 16 + 8 * half) * M_TOT + bn * 16 + l;
#pragma unroll
  for (int v = 0; v < 8; ++v) out[(size_t)v * M_TOT] = c[v];
}

__device__ __forceinline__ float tsig_arg(float x) {
  // sigma(x/T) with clamp: 1/(1+exp(clip(-x/T,-50,50)))
  float e = fminf(fmaxf(-INV_T * x, -50.0f), 50.0f);
  return 1.0f / (1.0f + __expf(e));
}

// ---------------------------------------------------------------------------
// Kernel 2: one thread per (j,k).  score_rank: 320-term sigmoid sum where
// score[i,j] is wave-uniform (consecutive lanes share j -> scalar-path loads).
// cider_rank collapsed to 64 terms using the KxK block structure of cbig.
// Ratio min/max, then LDS block reduction to one partial per block.
// ---------------------------------------------------------------------------
__global__ __launch_bounds__(256) void daa_rank_reduce(const float* __restrict__ S,
                                                       const float* __restrict__ C,
                                                       float* __restrict__ partials) {
  const int tid = blockIdx.x * 256 + threadIdx.x;   // tid in [0, 320*320)
  const int j   = tid / M_TOT;
  const int k   = tid - j * M_TOT;
  const int j5  = j / K_CAP;
  const int k5  = k / K_CAP;

  const float s_jk = S[(size_t)j * M_TOT + k];
  const float c_jk = C[j5 * N_IMG + k5];

  // score_rank_all[j,k] = 1 + sum_{i != k} sigma((S[i,j]-S[j,k])/T)
  float srank = 1.0f;
#pragma unroll 4
  for (int i = 0; i < M_TOT; ++i) {
    float t = tsig_arg(S[(size_t)i * M_TOT + j] - s_jk);
    srank += (i != k) ? t : 0.0f;
  }

  // cider_rank_all[j,k] = 1 + 5*sum_I sigma((C[I,j5]-C[j5,k5])/T)
  //                         -   sigma((C[k5,j5]-C[j5,k5])/T)
  float crank = 1.0f - tsig_arg(C[k5 * N_IMG + j5] - c_jk);
#pragma unroll 4
  for (int I = 0; I < N_IMG; ++I) {
    crank += 5.0f * tsig_arg(C[I * N_IMG + j5] - c_jk);
  }

  const float mn    = fminf(srank, crank);
  const float mx    = fmaxf(srank, crank);
  const float ratio = mn / mx;

  __shared__ float red[256];
  red[threadIdx.x] = ratio;
  __syncthreads();
#pragma unroll
  for (int s = 128; s > 0; s >>= 1) {
    if (threadIdx.x < (unsigned)s) red[threadIdx.x] += red[threadIdx.x + s];
    __syncthreads();
  }
  if (threadIdx.x == 0) partials[blockIdx.x] = red[0];
}

// ---------------------------------------------------------------------------
// Kernel 3: final reduction of 400 partials -> out = 1 - ASP
// ---------------------------------------------------------------------------
__global__ __launch_bounds__(512) void daa_finalize(const float* __restrict__ partials,
                                                    int n_partials,
                                                    float* __restrict__ out) {
  __shared__ float red[512];
  float v = (threadIdx.x < (unsigned)n_partials) ? partials[threadIdx.x] : 0.0f;
  red[threadIdx.x] = v;
  __syncthreads();
#pragma unroll
  for (int s = 256; s > 0; s >>= 1) {
    if (threadIdx.x < (unsigned)s) red[threadIdx.x] += red[threadIdx.x + s];
    __syncthreads();
  }
  if (threadIdx.x == 0)
    out[0] = 1.0f - red[0] / (float)((size_t)M_TOT * (size_t)M_TOT);
}

extern "C" void kernel_launch(void* const* d_in, const int* in_sizes, int n_in,
                              void* d_out, int out_size, void* d_ws, size_t ws_size,
                              hipStream_t stream) {
  const float* input1 = (const float*)d_in[0];   // (64,5,512) f32 — input2 unused
  const float* cider  = (const float*)d_in[2];   // (64,64) f32
  float* out = (float*)d_out;

  // workspace: [0, 320*320*4) = score matrix; then 400 partial sums
  float* score    = (float*)d_ws;
  float* partials = (float*)((char*)d_ws + (size_t)M_TOT * M_TOT * sizeof(float));

  // 1) score = A A^T via f32 WMMA (20x20 tiles of 16x16, one wave each)
  daa_score_gemm<<<dim3(M_TOT / 16, M_TOT / 16), 32, 0, stream>>>(input1, score);

  // 2) rank sums + ratio + block reduction (320*320 threads = 400 blocks x 256)
  const int n_blocks = (M_TOT * M_TOT) / 256;    // 400
  daa_rank_reduce<<<n_blocks, 256, 0, stream>>>(score, cider, partials);

  // 3) final scalar
  daa_finalize<<<1, 512, 0, stream>>>(partials, n_blocks, out);
}